// AvgPool2d_85478439125864
// MI455X (gfx1250) — compile-verified
//
#include <hip/hip_runtime.h>

typedef __attribute__((ext_vector_type(2))) float v2f;
typedef __attribute__((ext_vector_type(4))) float v4f;
typedef __attribute__((ext_vector_type(8))) float v8f;

#define HH   512
#define WWID 512
#define KWIN 16
#define NROW 48    // input rows staged per WG: 32 output rows need a 47-row window
#define XST  524   // xs LDS stride: 524 % 64 = 12 -> conflict-free b64 A-frag loads
#define YST  50    // per-wave y-tile stride: 50 % 64 -> 16 distinct even bank pairs

// window start with edge clamping: equivalent to reference cumsum + edge-pad
__device__ __forceinline__ int win_start(int i) {
    int s = i - 7;
    s = s < 0 ? 0 : s;
    return s > (HH - KWIN) ? (HH - KWIN) : s;   // clamp to 496
}

// 1.0f iff 0 <= off < 16, via single unsigned compare
__device__ __forceinline__ float band(int off) {
    return ((unsigned)off < (unsigned)KWIN) ? 1.0f : 0.0f;
}

// out = (1/256) * Aband * x * Aband^T per (n,c) slice, fused per 32-row stripe.
// blockIdx.x: 32-row output stripe (0..15), blockIdx.y: slice n*c (0..255)
__global__ __launch_bounds__(256)
void avgpool16x16_wmma(const float* __restrict__ x, float* __restrict__ out) {
    __shared__ float xs[NROW * XST];      // input rows r0..r0+47 (row-clamped) + zeroed pad
    __shared__ float ysw[8 * 16 * YST];   // per-wave transposed column-sum tile y[j_local][r_local]

    const int tid  = threadIdx.x;
    const int wave = tid >> 5;
    const int lane = tid & 31;
    const int l15  = lane & 15;
    const int hi   = lane >> 4;           // K offset +2 for lanes 16-31 in f32 WMMA frags

    const int slice = blockIdx.y;
    const int i0 = blockIdx.x << 5;       // 32 output rows per WG
    const int r0 = win_start(i0);

    const float* __restrict__ xin  = x   + (size_t)slice * (HH * WWID);
    float* __restrict__       xout = out + (size_t)slice * (HH * WWID);

    // ---- Step A: coalesced b128 load of 48 rows into LDS (rows clamped at 511) ----
    #pragma unroll
    for (int it = 0; it < 24; ++it) {
        int v   = tid + (it << 8);        // float4 index, 6144 total
        int row = v >> 7;                 // 128 float4 per row
        int col = (v & 127) << 2;
        int r   = r0 + row; r = (r > HH - 1) ? (HH - 1) : r;
        *(v4f*)&xs[row * XST + col] = *(const v4f*)(xin + r * WWID + col);
    }
    if (tid < NROW) {                     // zero stride padding: (band==0)*junk must stay 0
        for (int c = WWID; c < XST; ++c) xs[tid * XST + c] = 0.0f;
    }
    __syncthreads();

    float* myy = &ysw[wave * (16 * YST)];

    // ---- hoisted step-C band A-frags for the two 16-row output tiles (jt-invariant) ----
    const int kb1 = (win_start(i0 + 16) - r0) & ~1;   // even K base of tile 1 window
    v2f aC0[8], aC1[8];
    {
        const int si0 = win_start(i0 + l15);
        const int si1 = win_start(i0 + 16 + l15);
        #pragma unroll
        for (int kt = 0; kt < 8; ++kt) {
            int kk = (kt << 2) + (hi << 1);
            aC0[kt].x = band(r0 + kk - si0);
            aC0[kt].y = band(r0 + kk + 1 - si0);
            aC1[kt].x = band(r0 + kb1 + kk - si1);
            aC1[kt].y = band(r0 + kb1 + kk + 1 - si1);
        }
    }
    // ---- hoisted interior step-B band B-frags (identical for jt = 1..30) ----
    v2f bInt[8];
    #pragma unroll
    for (int kt = 0; kt < 8; ++kt) {
        int kk = (kt << 2) + (hi << 1);   // interior: c0 - sj = kk - 1 - l15
        bInt[kt].x = band(kk - 1 - l15);
        bInt[kt].y = band(kk - l15);
    }

    for (int jt = wave; jt < 32; jt += 8) {
        const int j0  = jt << 4;
        const int cb0 = win_start(j0) & ~1;   // even K base; [cb0, cb0+31] covers all 16 col-bands

        v2f bb[8];
        if (jt == 0 || jt == 31) {            // wave-uniform branch: edge tiles recompute bands
            const int sj = win_start(j0 + l15);
            #pragma unroll
            for (int kt = 0; kt < 8; ++kt) {
                int c0 = cb0 + (kt << 2) + (hi << 1);
                bb[kt].x = band(c0 - sj);
                bb[kt].y = band(c0 + 1 - sj);
            }
        } else {
            #pragma unroll
            for (int kt = 0; kt < 8; ++kt) bb[kt] = bInt[kt];
        }

        // ---- Step B: y[r][j] = sum_c x[r][c]*band_j(c) for 48 rows, stored transposed ----
        #pragma unroll
        for (int rb = 0; rb < 3; ++rb) {
            v8f acc = {};
            #pragma unroll
            for (int kt = 0; kt < 8; ++kt) {
                int kk = (kt << 2) + (hi << 1);
                v2f a = *(const v2f*)&xs[((rb << 4) + l15) * XST + cb0 + kk];  // aligned b64
                acc = __builtin_amdgcn_wmma_f32_16x16x4_f32(false, a, false, bb[kt],
                                                            (short)0, acc, false, false);
            }
            #pragma unroll
            for (int p = 0; p < 8; ++p)        // D[m=r_local][n=j_local] -> y[j][r] transposed
                myy[l15 * YST + (rb << 4) + p + (hi << 3)] = acc[p];
        }
        // wave-private LDS; same-wave DS ops are in-order, compiler emits s_wait_dscnt

        // ---- Step C: two 16-row output tiles from the shared y rows ----
        v8f acc0 = {}, acc1 = {};
        #pragma unroll
        for (int kt = 0; kt < 8; ++kt) {
            int kk = (kt << 2) + (hi << 1);
            v2f b0 = *(const v2f*)&myy[l15 * YST + kk];          // tile0: K window [0,31]
            acc0 = __builtin_amdgcn_wmma_f32_16x16x4_f32(false, aC0[kt], false, b0,
                                                         (short)0, acc0, false, false);
            v2f b1 = *(const v2f*)&myy[l15 * YST + kb1 + kk];    // tile1: K window [kb1, kb1+31]
            acc1 = __builtin_amdgcn_wmma_f32_16x16x4_f32(false, aC1[kt], false, b1,
                                                         (short)0, acc1, false, false);
        }
        #pragma unroll
        for (int p = 0; p < 8; ++p) {
            int orow0 = i0 + p + (hi << 3);
            xout[orow0 * WWID + j0 + l15] = acc0[p] * (1.0f / 256.0f);
            int orow1 = i0 + 16 + p + (hi << 3);
            xout[orow1 * WWID + j0 + l15] = acc1[p] * (1.0f / 256.0f);
        }
    }
}

extern "C" void kernel_launch(void* const* d_in, const int* in_sizes, int n_in,
                              void* d_out, int out_size, void* d_ws, size_t ws_size,
                              hipStream_t stream) {
    (void)in_sizes; (void)n_in; (void)out_size; (void)d_ws; (void)ws_size;
    const float* x = (const float*)d_in[0];
    float* out = (float*)d_out;
    // 4*64 = 256 slices, 512/32 = 16 row stripes per slice
    dim3 grid(16, 256);
    avgpool16x16_wmma<<<grid, dim3(256), 0, stream>>>(x, out);
}